// OGNN_layer_10548439679296
// MI455X (gfx1250) — compile-verified
//
#include <hip/hip_runtime.h>
#include <hip/hip_bf16.h>

// ---------------------------------------------------------------------------
// OGNN layer for MI455X (gfx1250, wave32):
//   hamB  = octonion_expand(weight) in WMMA B-fragment order   (kernel 1)
//   support = input @ hamilton  (V_WMMA_F32_16X16X4_F32)       (kernel 2)
//   agg   = segment_sum(val * support[col])  (fp32 HW atomics) (kernels 3+4)
//   out   = tanh(BN(agg))                                      (kernels 5+6)
// agg lives directly in d_out; d_ws holds hamB + support + stats.
//
// Roofline: SpMM dominates (800K edges * 512B gather + 102M f32 atomics, all
// L2-resident at 192MB). GEMM is 1.64 GFLOP -> f32 WMMA keeps full precision
// at negligible cost. HBM floor ~60MB @ 23.3TB/s.
// ---------------------------------------------------------------------------

#define D 128            // D_IN == D_OUT == 128
#define D4 32            // D/4

typedef float v2f __attribute__((ext_vector_type(2)));
typedef float v8f __attribute__((ext_vector_type(8)));

// (component, sign) tables indexed [col_block][row_block], matching _OCT_TABLE.
__device__ __constant__ int OCT_COMP[8][8] = {
  {0,1,2,3,4,5,6,7}, {1,0,3,2,5,4,7,6}, {2,3,0,1,6,7,4,5}, {3,2,1,0,7,6,5,4},
  {4,5,6,7,0,1,2,3}, {5,4,7,6,1,0,3,2}, {6,7,4,5,2,3,0,1}, {7,6,5,4,3,2,1,0}};
__device__ __constant__ float OCT_SIGN[8][8] = {
  { 1,-1,-1,-1,-1,-1,-1,-1}, { 1, 1,-1, 1,-1, 1, 1,-1},
  { 1, 1, 1,-1,-1,-1, 1, 1}, { 1,-1, 1, 1,-1, 1,-1, 1},
  { 1, 1, 1, 1, 1,-1,-1,-1}, { 1,-1, 1,-1, 1, 1, 1,-1},
  { 1,-1,-1, 1, 1,-1, 1, 1}, { 1, 1,-1,-1, 1, 1,-1, 1}};

__device__ __forceinline__ void atomAddF(float* p, float v) {
  __hip_atomic_fetch_add(p, v, __ATOMIC_RELAXED, __HIP_MEMORY_SCOPE_AGENT);
}

// ---- Kernel 1: expand weight[16,128] into hamilton, stored directly in
// V_WMMA_F32_16X16X4_F32 B-fragment order: hf[((kk*8+nt)*32+lane)*2+e] holds
// hamilton[K][nt*16 + lane%16] with K = 4*kk + 2*(lane/16) + e.
__global__ void k_build_hamB(const float* __restrict__ w, float* __restrict__ hf) {
  int f = blockIdx.x * 256 + threadIdx.x;       // 16384 threads total
  int e    = f & 1;
  int lane = (f >> 1) & 31;
  int nt   = (f >> 6) & 7;
  int kk   = f >> 9;
  int m = lane & 15, h = lane >> 4;
  int K = kk * 4 + 2 * h + e;                   // hamilton row index
  int ri = K >> 4, r = K & 15;                  // row block / row within block
  // hamilton[K][nt*16+m] = sign[nt][ri] * weight[r][comp[nt][ri]*16 + m]
  hf[f] = OCT_SIGN[nt][ri] * w[r * D + OCT_COMP[nt][ri] * 16 + m];
}

// ---- Kernel 2: support = input @ hamilton via WMMA -------------------------
// Block = 256 threads = 8 waves; each block computes a 128x128 output slab,
// each wave a 16-row strip. B fragments come from LDS as single aligned
// ds_load_b64's (conflict-free: lane*8B consecutive); A slab is staged with a
// padded stride (banks 4*m, +2 for upper half -> disjoint).
#define LDA 132   // sA row stride in floats
__global__ void __launch_bounds__(256)
k_gemm_wmma(const float* __restrict__ input, const float* __restrict__ hamB,
            float* __restrict__ support, int n) {
  extern __shared__ float smem[];
  float* sB = smem;                 // 16384 floats, fragment order
  float* sA = smem + D * D;         // [128][LDA]
  const int tid = threadIdx.x;
  const int rowBase = blockIdx.x * 128;

  // Stage B fragments (coalesced float4 linear copy).
  {
    float4* sB4 = (float4*)sB;
    const float4* h4 = (const float4*)hamB;
    for (int i = tid; i < D * D4; i += 256) sB4[i] = h4[i];
  }
  // Stage 128 input rows (zero-padded past n).
  {
    float4* sA4 = (float4*)sA;
    const float4* in4 = (const float4*)input;
    for (int i = tid; i < D * D4; i += 256) {
      int rr = i >> 5, cc = i & 31;
      int g = rowBase + rr;
      float4 v = make_float4(0.f, 0.f, 0.f, 0.f);
      if (g < n) v = in4[(size_t)g * D4 + cc];
      sA4[rr * (LDA / 4) + cc] = v;
    }
  }
  __syncthreads();

  const int wave = tid >> 5, lane = tid & 31;
  const int m = lane & 15, h = lane >> 4;
  const int lr = wave * 16 + m;                   // A row for this lane
  const v2f* sBf = (const v2f*)sB;                // fragment (kk,nt) at (kk*8+nt)*32+lane

  v8f acc[8];
#pragma unroll
  for (int t = 0; t < 8; ++t) acc[t] = v8f{0, 0, 0, 0, 0, 0, 0, 0};

  for (int kk = 0; kk < 32; ++kk) {
    // A: K = 4*kk + 2*half + elem -> contiguous pair, 8B aligned.
    v2f a = *(const v2f*)&sA[lr * LDA + kk * 4 + 2 * h];
    const int base = kk * 8 * 32 + lane;
#pragma unroll
    for (int nt = 0; nt < 8; ++nt) {
      v2f b = sBf[base + nt * 32];
      acc[nt] = __builtin_amdgcn_wmma_f32_16x16x4_f32(
          false, a, false, b, (short)0, acc[nt], false, false);
    }
  }

  // D layout: M = 8*half + vgpr, N = lane%16.
#pragma unroll
  for (int nt = 0; nt < 8; ++nt)
#pragma unroll
    for (int v = 0; v < 8; ++v) {
      int grow = rowBase + wave * 16 + 8 * h + v;
      if (grow < n) support[(size_t)grow * D + nt * 16 + m] = acc[nt][v];
    }
}

// ---- Kernel 3: zero the accumulator (d_out) and the stats buffer -----------
__global__ void k_zero(float4* __restrict__ out4, int n4, float4* __restrict__ stats4) {
  int i = blockIdx.x * 256 + threadIdx.x;
  float4 z = make_float4(0.f, 0.f, 0.f, 0.f);
  if (i < n4) out4[i] = z;
  if (i < 64) stats4[i] = z;   // 256 floats: [0,128)=sum, [128,256)=sumsq
}

// ---- Kernel 4: SpMM scatter-add. One wave per edge, one float4 per lane. ---
__global__ void __launch_bounds__(256)
k_spmm(const int* __restrict__ adj_row, const int* __restrict__ adj_col,
       const float* __restrict__ adj_val, const float* __restrict__ support,
       float* __restrict__ out, int nEdges) {
  int e = blockIdx.x * 8 + (threadIdx.x >> 5);
  if (e >= nEdges) return;
  const int lane = threadIdx.x & 31;
  const int c = adj_col[e];
  const int r = adj_row[e];
  const float v = adj_val[e];
  float4 s = ((const float4*)support)[(size_t)c * D4 + lane];  // 512B/wave, L2-resident
  float* dst = out + (size_t)r * D + lane * 4;
  atomAddF(dst + 0, v * s.x);
  atomAddF(dst + 1, v * s.y);
  atomAddF(dst + 2, v * s.z);
  atomAddF(dst + 3, v * s.w);
}

// ---- Kernel 5: per-column sum / sumsq (thread <-> column => coalesced) -----
__global__ void __launch_bounds__(256)
k_stats(const float* __restrict__ out, float* __restrict__ stats, int n) {
  __shared__ float ssum[256], ssq[256];
  const int tid = threadIdx.x;
  const int j = tid & 127;
  float sum = 0.f, sq = 0.f;
  for (int r = blockIdx.x * 2 + (tid >> 7); r < n; r += gridDim.x * 2) {
    float x = out[(size_t)r * D + j];
    sum += x; sq += x * x;
  }
  ssum[tid] = sum; ssq[tid] = sq;
  __syncthreads();
  if (tid < 128) {
    atomAddF(&stats[j],     ssum[tid] + ssum[tid + 128]);
    atomAddF(&stats[D + j], ssq[tid]  + ssq[tid + 128]);
  }
}

// ---- Kernel 6: BatchNorm (biased stats) + tanh, in place on d_out ----------
__global__ void k_bn_tanh(float* __restrict__ out, const float* __restrict__ stats,
                          const float* __restrict__ gamma, const float* __restrict__ beta,
                          int total, float invN) {
  int i = blockIdx.x * 256 + threadIdx.x;
  if (i >= total) return;
  int j = i & 127;
  float mean = stats[j] * invN;
  float var  = stats[D + j] * invN - mean * mean;
  float rstd = rsqrtf(var + 1e-5f);
  float x = (out[i] - mean) * rstd * gamma[j] + beta[j];
  out[i] = tanhf(x);
}

// ---------------------------------------------------------------------------
extern "C" void kernel_launch(void* const* d_in, const int* in_sizes, int n_in,
                              void* d_out, int out_size, void* d_ws, size_t ws_size,
                              hipStream_t stream) {
  const float* input   = (const float*)d_in[0];
  const int*   adj_row = (const int*)  d_in[1];
  const int*   adj_col = (const int*)  d_in[2];
  const float* adj_val = (const float*)d_in[3];
  const float* weight  = (const float*)d_in[4];
  const float* gamma   = (const float*)d_in[5];
  const float* beta    = (const float*)d_in[6];
  float* out = (float*)d_out;

  const int N = in_sizes[0] / D;       // 50000 nodes
  const int E = in_sizes[1];           // 800000 edges

  // Workspace layout (floats): hamB[16384] | support[N*128] | stats[256]
  float* hamB    = (float*)d_ws;
  float* support = hamB + D * D;
  float* stats   = support + (size_t)N * D;

  // 1) hamilton expansion directly into B-fragment order (16384 elems)
  k_build_hamB<<<64, 256, 0, stream>>>(weight, hamB);

  // 2) WMMA GEMM: 128 rows per block, ~130 KB dynamic LDS
  const int gemmBlocks = (N + 127) / 128;
  const size_t ldsBytes = (size_t)(D * D + D * LDA) * sizeof(float);
  k_gemm_wmma<<<gemmBlocks, 256, ldsBytes, stream>>>(input, hamB, support, N);

  // 3) zero accumulator + stats
  const int n4 = N * D4;
  k_zero<<<(n4 + 255) / 256, 256, 0, stream>>>((float4*)out, n4, (float4*)stats);

  // 4) edge scatter-add (wave32 per edge)
  k_spmm<<<(E + 7) / 8, 256, 0, stream>>>(adj_row, adj_col, adj_val, support, out, E);

  // 5) column statistics
  k_stats<<<512, 256, 0, stream>>>(out, stats, N);

  // 6) normalize + tanh
  const int total = N * D;
  k_bn_tanh<<<(total + 255) / 256, 256, 0, stream>>>(out, stats, gamma, beta,
                                                     total, 1.0f / (float)N);
}